// test_70480413327683
// MI455X (gfx1250) — compile-verified
//
#include <hip/hip_runtime.h>

typedef __attribute__((ext_vector_type(2))) float v2f;
typedef __attribute__((ext_vector_type(8))) float v8f;

#define NEG_SLOPE 0.01f

__device__ __forceinline__ float leaky(float v) {
    return v >= 0.0f ? v : NEG_SLOPE * v;
}

__device__ __forceinline__ float sigmoidf(float z) {
    return 1.0f / (1.0f + __expf(-z));
}

// One wave32 handles 16 coarse voxels (same b,d,e,f; g = lane%16).
// Stage1: D(16x16) = W1pad(16x48) * Y0(48x16) via 12x v_wmma_f32_16x16x4_f32
// Stage2: 3 tiles of D(16x16) = Wt_tile(16x8pad) * Y1(8pad x16) via 2 WMMAs each.
__global__ __launch_bounds__(256)
void fused_conv4d_wmma(const float* __restrict__ x,
                       const float* __restrict__ w0,
                       const float* __restrict__ b0,
                       const float* __restrict__ w1,
                       const float* __restrict__ b1,
                       const float* __restrict__ wt,
                       const float* __restrict__ bt,
                       float* __restrict__ out)
{
    // per-wave LDS slice: xblk[16 pos][stride 17] (272) + y1buf[6][16] (96)
    __shared__ float lds[8][368];

    const int tid  = threadIdx.x;
    const int lane = tid & 31;
    const int warp = tid >> 5;
    const int n    = lane & 15;   // voxel column (N dim), also g-coarse
    const int half = lane >> 4;   // upper/lower half of wave
    const int m    = n;           // A-matrix row for this lane

    float* xblk  = &lds[warp][0];
    float* y1buf = &lds[warp][272];

    // global wave id -> (b,d,e,f) coarse coordinates (D=32 -> 16^4 per batch)
    const int W = blockIdx.x * 8 + warp;
    const int b = W >> 12;
    const int d = (W >> 8) & 15;
    const int e = (W >> 4) & 15;
    const int f = W & 15;

    // ---- stage x->LDS: each lane loads 4 float2 (s=0,1 pairs), fully coalesced
    #pragma unroll
    for (int i = 0; i < 4; ++i) {
        const int pos8 = 4 * half + i;          // p*4 + q*2 + r
        const int p = (pos8 >> 2) & 1;
        const int q = (pos8 >> 1) & 1;
        const int r = pos8 & 1;
        const size_t idx =
            ((((size_t)b * 32 + 2 * d + p) * 32 + 2 * e + q) * 32 + 2 * f + r) * 32
            + 2 * n;
        const float2 xv = *reinterpret_cast<const float2*>(x + idx);
        xblk[(pos8 * 2 + 0) * 17 + n] = xv.x;
        xblk[(pos8 * 2 + 1) * 17 + n] = xv.y;
    }

    // ---- preload tiny weights into registers (all L2/WGP$ hits)
    float w0r[3], b0r[3];
    #pragma unroll
    for (int c = 0; c < 3; ++c) { w0r[c] = w0[c]; b0r[c] = b0[c]; }

    // A-operand for stage 1: W1pad[m, kk], kk = 4k + 2*half + j (ISA 16x4 layout)
    float a1v[12][2];
    #pragma unroll
    for (int k = 0; k < 12; ++k) {
        #pragma unroll
        for (int j = 0; j < 2; ++j) {
            const int kk = 4 * k + 2 * half + j;
            a1v[k][j] = (m < 6) ? w1[m * 48 + kk] : 0.0f;
        }
    }

    __syncthreads();

    // ---- stage 1: 12 chained f32 WMMAs over K=48 (c*16 + pos)
    v8f acc = {0.f, 0.f, 0.f, 0.f, 0.f, 0.f, 0.f, 0.f};
    #pragma unroll
    for (int k = 0; k < 12; ++k) {
        const int kb = 4 * k + 2 * half;
        v2f a, bb;
        a[0] = a1v[k][0];
        a[1] = a1v[k][1];
        #pragma unroll
        for (int j = 0; j < 2; ++j) {
            const int kk  = kb + j;
            const int c   = kk >> 4;
            const int pos = kk & 15;
            // stage-0 pointwise conv + LeakyReLU fused into B-operand build
            bb[j] = leaky(w0r[c] * xblk[pos * 17 + n] + b0r[c]);
        }
        acc = __builtin_amdgcn_wmma_f32_16x16x4_f32(false, a, false, bb,
                                                    (short)0, acc, false, false);
    }

    // ---- stage-1 epilogue: lanes 0-15 hold rows M=0..7 -> channels o=0..5
    if (half == 0) {
        #pragma unroll
        for (int o = 0; o < 6; ++o) {
            y1buf[o * 16 + n] = leaky(acc[o] + b1[o]);
        }
    }
    __syncthreads();

    // ---- stage 2: three M=16 tiles (output channel t), K=6 padded to 8
    #pragma unroll
    for (int t = 0; t < 3; ++t) {
        v8f acc2 = {0.f, 0.f, 0.f, 0.f, 0.f, 0.f, 0.f, 0.f};
        #pragma unroll
        for (int ks = 0; ks < 2; ++ks) {
            const int kb = 4 * ks + 2 * half;
            v2f a, bb;
            #pragma unroll
            for (int j = 0; j < 2; ++j) {
                const int i = kb + j;               // input channel (K index)
                a[j]  = (i < 6) ? wt[i * 48 + t * 16 + m] : 0.0f;
                bb[j] = (i < 6) ? y1buf[i * 16 + n] : 0.0f;
            }
            acc2 = __builtin_amdgcn_wmma_f32_16x16x4_f32(false, a, false, bb,
                                                         (short)0, acc2, false, false);
        }
        const float btv = bt[t];
        // D rows: pos = j + 8*half; pair (j=2u, 2u+1) = contiguous s=0/1 -> float2
        #pragma unroll
        for (int u = 0; u < 4; ++u) {
            const int p = half;
            const int q = (u >> 1) & 1;
            const int r = u & 1;
            float2 o2;
            o2.x = sigmoidf(acc2[2 * u]     + btv);
            o2.y = sigmoidf(acc2[2 * u + 1] + btv);
            const size_t oidx =
                (((((size_t)b * 3 + t) * 32 + 2 * d + p) * 32 + 2 * e + q) * 32
                 + 2 * f + r) * 32 + 2 * n;
            *reinterpret_cast<float2*>(out + oidx) = o2;
        }
    }
}

extern "C" void kernel_launch(void* const* d_in, const int* in_sizes, int n_in,
                              void* d_out, int out_size, void* d_ws, size_t ws_size,
                              hipStream_t stream) {
    const float* x  = (const float*)d_in[0];
    const float* w0 = (const float*)d_in[1];
    const float* b0 = (const float*)d_in[2];
    const float* w1 = (const float*)d_in[3];
    const float* b1 = (const float*)d_in[4];
    const float* wt = (const float*)d_in[5];
    const float* bt = (const float*)d_in[6];
    float* out = (float*)d_out;

    const int Bsz = in_sizes[0] / (32 * 32 * 32 * 32);   // = 4
    const int totalWaves = Bsz * 16 * 16 * 16 * 16 / 16; // 16 voxels per wave
    const int blocks = totalWaves / 8;                   // 8 waves per block

    fused_conv4d_wmma<<<blocks, 256, 0, stream>>>(x, w0, b0, w1, b1, wt, bt, out);
}